// DisparityRegression_26800595927729
// MI455X (gfx1250) — compile-verified
//
#include <hip/hip_runtime.h>
#include <math.h>

#define D        192                 // disparities per row
#define RPW      8                   // rows per wave (one TDM tile)
#define LPR      4                   // lanes cooperating on one row
#define EPL      (D / LPR)           // 48 elements per lane
#define WAVES    8                   // waves per 256-thread block
#define ROWS_PB  (RPW * WAVES)       // 64 rows per block
#define SROW     196                 // LDS row stride in floats (768B + 16B pad, via TDM iterate)

typedef int v4i __attribute__((ext_vector_type(4)));
typedef int v8i __attribute__((ext_vector_type(8)));

// jax.lax.top_k ordering: larger value wins; ties broken by LOWER index.
__device__ __forceinline__ bool tk_better(float v, int i, float w, int j) {
  return (v > w) || ((v == w) && (i < j));
}

__global__ __launch_bounds__(256) void disparity_top2_kernel(
    const float* __restrict__ cost, float* __restrict__ out, int nrows) {
  // 64 rows * 196 floats * 4B = 50176 B staged per block (320 KB/WGP available)
  __shared__ __align__(16) float smem[WAVES * RPW * SROW];

  const int wave = threadIdx.x >> 5;       // wave32
  const int lane = threadIdx.x & 31;
  const int row0 = blockIdx.x * ROWS_PB + wave * RPW;
  if (row0 >= nrows) return;               // uniform per wave (nrows % 64 == 0)

  float* wbase = &smem[wave * RPW * SROW];
  const char* g0 = (const char*)(cost + (size_t)row0 * D);

  // ---- TDM: one tensor_load_to_lds moves this wave's 8 rows (6 KB) ----
  // D# (ISA 8.3-8.6): 1-D tile of 192 fp32, iterate 8x; global advances 192
  // elements (contiguous rows), LDS advances 196 elements -> 16B row padding
  // so the ds_load_b128 read pattern below hits each 4-bank group exactly
  // twice (minimal for 128-bit LDS reads).
  {
    unsigned long long ga = (unsigned long long)g0;        // wave-uniform
    unsigned galo = __builtin_amdgcn_readfirstlane((unsigned)ga);
    unsigned gahi = __builtin_amdgcn_readfirstlane((unsigned)(ga >> 32));
    unsigned lds  = __builtin_amdgcn_readfirstlane(
                        (unsigned)(size_t)wbase);          // LDS byte offset
    v4i d0 = { 1,                                          // count=1 descriptor
               (int)lds,                                   // lds_addr
               (int)galo,                                  // global_addr[31:0]
               (int)((gahi & 0x01FFFFFFu) | 0x80000000u) };// ga[56:32] | type=2
    v8i d1 = { (2 << 16) | (1 << 19),                      // data_size=4B, iterate_enable
               192 << 16,                                  // tensor_dim0 = 192
               8 << 16,                                    // tensor_dim1 = 8
               192 << 16,                                  // tile_dim0 = 192 (1-D tile)
               0,                                          // tile_dim1/2 = 0
               192, 0, 0 };                                // tensor_dim0_stride = 192
    v4i d2 = { 0,                                          // tensor_dim2
               196,                                        // lds_addr_increment
               192,                                        // global_addr_increment
               7 << 16 };                                  // iterate_count = 7 (8 iters)
    v4i d3 = { 0, 0, 0, 0 };
    asm volatile("tensor_load_to_lds %0, %1, %2, %3"
                 :: "s"(d0), "s"(d1), "s"(d2), "s"(d3) : "memory");
  }
  // prime L2 for the next block's tile while the TDM copy is in flight
  __builtin_prefetch(g0 + (size_t)ROWS_PB * D * 4 + lane * 16, 0, 1);
  asm volatile("s_wait_tensorcnt 0x0" ::: "memory");

  // ---- branchless per-lane top-2 over 48 contiguous elements ----
  const int sub   = lane & (LPR - 1);      // 0..3 within the row
  const int r     = lane >> 2;             // which of this wave's 8 rows
  const float* rp = wbase + r * SROW + sub * EPL;
  const int ibase = sub * EPL;

  float v1 = -INFINITY, v2 = -INFINITY;
  int   k1 = 0x7fff, k2 = 0x7fff;          // local ordinals: inline-const selects
#pragma unroll
  for (int t = 0; t < EPL / 4; ++t) {
    float4 q = ((const float4*)rp)[t];     // ds_load_b128, 2 bank passes
    float e[4] = {q.x, q.y, q.z, q.w};
#pragma unroll
    for (int kk = 0; kk < 4; ++kk) {
      float v = e[kk];
      int   k = t * 4 + kk;                // 0..47: wave-uniform inline constant
      // strict '>' keeps the earliest index on exact ties (top_k semantics)
      bool  c = v > v1;
      float w = c ? v1 : v;                // loser of (v, v1)  -> v_min
      int   j = c ? k1 : k;
      v1 = c ? v : v1;                     // winner            -> v_max
      k1 = c ? k : k1;
      bool c2 = w > v2;
      v2 = c2 ? w : v2;                    //                   -> v_max
      k2 = c2 ? j : k2;
    }
  }
  int i1 = ibase + k1, i2 = ibase + k2;    // materialize global indices once

  // ---- merge sorted (top1, top2) pairs across the 4 lanes of this row ----
#pragma unroll
  for (int m = 1; m <= 2; m <<= 1) {
    float ov1 = __shfl_xor(v1, m, 32); int oi1 = __shfl_xor(i1, m, 32);
    float ov2 = __shfl_xor(v2, m, 32); int oi2 = __shfl_xor(i2, m, 32);
    bool  up = tk_better(ov1, oi1, v1, i1);
    float a1 = up ? ov1 : v1;  int b1 = up ? oi1 : i1;   // merged top
    float lx = up ? v1  : ov1; int li = up ? i1  : oi1;  // losing top
    float s2 = up ? ov2 : v2;  int t2 = up ? oi2 : i2;   // winner's second
    bool  ss = tk_better(lx, li, s2, t2);
    v2 = ss ? lx : s2;  i2 = ss ? li : t2;               // merged second
    v1 = a1;            i1 = b1;
  }

  // ---- softmax over the two candidates + disparity regression ----
  if (sub == 0) {                          // 8 lanes -> 8 consecutive rows
    float d  = v1 - v2;                    // >= 0: exp never blows up p2
    float p2 = 1.0f / (1.0f + expf(d));    // weight of the runner-up
    out[row0 + r] = (float)i1 + ((float)i2 - (float)i1) * p2;
  }
}

extern "C" void kernel_launch(void* const* d_in, const int* in_sizes, int n_in,
                              void* d_out, int out_size, void* d_ws, size_t ws_size,
                              hipStream_t stream) {
  const float* cost = (const float*)d_in[0];
  float* out = (float*)d_out;
  const int nrows  = in_sizes[0] / D;                    // 4*131072 = 524288
  const int blocks = (nrows + ROWS_PB - 1) / ROWS_PB;    // 8192 blocks
  disparity_top2_kernel<<<blocks, 256, 0, stream>>>(cost, out, nrows);
}